// RGCN_35485019799713
// MI455X (gfx1250) — compile-verified
//
#include <hip/hip_runtime.h>

typedef __attribute__((ext_vector_type(2))) float v2f;
typedef __attribute__((ext_vector_type(8))) float v8f;

#define N_NODES 50000
#define IN_DIM  128
#define H1_DIM  64
#define H2_DIM  32
#define OUT_DIM 64
#define N_REL   8
#define N_TILES (N_NODES / 16)   // 3125 exactly

static __device__ __forceinline__ v8f wmma4(v2f a, v2f b, v8f c) {
    // D = A(16x4) * B(4x16) + C(16x16), all f32, wave32
    return __builtin_amdgcn_wmma_f32_16x16x4_f32(
        /*neg_a=*/false, a, /*neg_b=*/false, b,
        /*c_mod=*/(short)0, c, /*reuse_a=*/false, /*reuse_b=*/false);
}

// ---------------------------------------------------------------------------
// Kernel A: encoder MLP + per-relation transform + self-loop.
// One wave32 per 16-node tile; 4 waves per block. Each wave owns its LDS slice
// so no inter-wave barriers are required (in-wave DS ops are in-order).
// ---------------------------------------------------------------------------
__global__ void __launch_bounds__(128)
rgcn_node_kernel(const float* __restrict__ X,
                 const float* __restrict__ W1, const float* __restrict__ b1,
                 const float* __restrict__ W2, const float* __restrict__ b2,
                 const float* __restrict__ Wrel,
                 const float* __restrict__ Wloop,
                 const float* __restrict__ bias,
                 float* __restrict__ hW,    // [R, N, OUT]
                 float* __restrict__ acc)   // [N, OUT]  (self-loop + bias)
{
    __shared__ float sX [4][16][132];  // row pad 132 (mod 64 == 4): conflict-free A reads
    __shared__ float sH1[4][16][68];   // pad 68
    __shared__ float sH2[4][16][36];   // pad 36

    const int lane = threadIdx.x & 31;
    const int w    = threadIdx.x >> 5;
    const int tile = blockIdx.x * 4 + w;
    if (tile >= N_TILES) return;       // wave-uniform: EXEC stays all-1s for WMMA

    const int half = lane >> 4;        // 0: K even pair / rows 0-7 ; 1: K odd pair / rows 8-15
    const int m16  = lane & 15;
    const int base = tile * 16;

    // ---- stage X tile into LDS (coalesced) ----------------------------------
    for (int idx = lane; idx < 16 * IN_DIM; idx += 32) {
        int r = idx >> 7, c = idx & 127;
        sX[w][r][c] = X[(size_t)(base + r) * IN_DIM + c];
    }

    // ---- stage 1: H1 = relu(X @ W1 + b1)   [16 x 64] ------------------------
    {
        v8f c0 = {}, c1 = {}, c2 = {}, c3 = {};
        for (int k = 0; k < IN_DIM; k += 4) {
            v2f a;
            a.x = sX[w][m16][k + 2 * half + 0];
            a.y = sX[w][m16][k + 2 * half + 1];
            const float* wk = W1 + (size_t)(k + 2 * half) * H1_DIM;
            v2f b;
            b.x = wk[ 0 + m16]; b.y = wk[H1_DIM +  0 + m16]; c0 = wmma4(a, b, c0);
            b.x = wk[16 + m16]; b.y = wk[H1_DIM + 16 + m16]; c1 = wmma4(a, b, c1);
            b.x = wk[32 + m16]; b.y = wk[H1_DIM + 32 + m16]; c2 = wmma4(a, b, c2);
            b.x = wk[48 + m16]; b.y = wk[H1_DIM + 48 + m16]; c3 = wmma4(a, b, c3);
        }
        v8f cc[4] = {c0, c1, c2, c3};
        #pragma unroll
        for (int ct = 0; ct < 4; ++ct) {
            float bv = b1[ct * 16 + m16];
            #pragma unroll
            for (int v = 0; v < 8; ++v) {
                float x = cc[ct][v] + bv;
                sH1[w][v + 8 * half][ct * 16 + m16] = fmaxf(x, 0.0f);
            }
        }
    }

    // ---- stage 2: H2 = relu(H1 @ W2 + b2)  [16 x 32] ------------------------
    {
        v8f d0 = {}, d1 = {};
        for (int k = 0; k < H1_DIM; k += 4) {
            v2f a;
            a.x = sH1[w][m16][k + 2 * half + 0];
            a.y = sH1[w][m16][k + 2 * half + 1];
            const float* wk = W2 + (size_t)(k + 2 * half) * H2_DIM;
            v2f b;
            b.x = wk[ 0 + m16]; b.y = wk[H2_DIM +  0 + m16]; d0 = wmma4(a, b, d0);
            b.x = wk[16 + m16]; b.y = wk[H2_DIM + 16 + m16]; d1 = wmma4(a, b, d1);
        }
        v8f dd[2] = {d0, d1};
        #pragma unroll
        for (int ct = 0; ct < 2; ++ct) {
            float bv = b2[ct * 16 + m16];
            #pragma unroll
            for (int v = 0; v < 8; ++v) {
                float x = dd[ct][v] + bv;
                sH2[w][v + 8 * half][ct * 16 + m16] = fmaxf(x, 0.0f);
            }
        }
    }

    // ---- stage 3: hW[r] = H2 @ W_rel[r]    [16 x 64] per relation -----------
    for (int r = 0; r < N_REL; ++r) {
        const float* Wr = Wrel + (size_t)r * H2_DIM * OUT_DIM;
        v8f e0 = {}, e1 = {}, e2 = {}, e3 = {};
        #pragma unroll
        for (int k = 0; k < H2_DIM; k += 4) {
            v2f a;
            a.x = sH2[w][m16][k + 2 * half + 0];
            a.y = sH2[w][m16][k + 2 * half + 1];
            const float* wk = Wr + (size_t)(k + 2 * half) * OUT_DIM;
            v2f b;
            b.x = wk[ 0 + m16]; b.y = wk[OUT_DIM +  0 + m16]; e0 = wmma4(a, b, e0);
            b.x = wk[16 + m16]; b.y = wk[OUT_DIM + 16 + m16]; e1 = wmma4(a, b, e1);
            b.x = wk[32 + m16]; b.y = wk[OUT_DIM + 32 + m16]; e2 = wmma4(a, b, e2);
            b.x = wk[48 + m16]; b.y = wk[OUT_DIM + 48 + m16]; e3 = wmma4(a, b, e3);
        }
        v8f ee[4] = {e0, e1, e2, e3};
        #pragma unroll
        for (int ct = 0; ct < 4; ++ct)
            #pragma unroll
            for (int v = 0; v < 8; ++v)
                hW[((size_t)r * N_NODES + base + v + 8 * half) * OUT_DIM
                   + ct * 16 + m16] = ee[ct][v];
    }

    // ---- stage 4: acc = H2 @ W_loop + bias  [16 x 64] -----------------------
    {
        v8f f0 = {}, f1 = {}, f2 = {}, f3 = {};
        #pragma unroll
        for (int k = 0; k < H2_DIM; k += 4) {
            v2f a;
            a.x = sH2[w][m16][k + 2 * half + 0];
            a.y = sH2[w][m16][k + 2 * half + 1];
            const float* wk = Wloop + (size_t)(k + 2 * half) * OUT_DIM;
            v2f b;
            b.x = wk[ 0 + m16]; b.y = wk[OUT_DIM +  0 + m16]; f0 = wmma4(a, b, f0);
            b.x = wk[16 + m16]; b.y = wk[OUT_DIM + 16 + m16]; f1 = wmma4(a, b, f1);
            b.x = wk[32 + m16]; b.y = wk[OUT_DIM + 32 + m16]; f2 = wmma4(a, b, f2);
            b.x = wk[48 + m16]; b.y = wk[OUT_DIM + 48 + m16]; f3 = wmma4(a, b, f3);
        }
        v8f ff[4] = {f0, f1, f2, f3};
        #pragma unroll
        for (int ct = 0; ct < 4; ++ct) {
            float bv = bias[ct * 16 + m16];
            #pragma unroll
            for (int v = 0; v < 8; ++v)
                acc[(size_t)(base + v + 8 * half) * OUT_DIM + ct * 16 + m16]
                    = ff[ct][v] + bv;
        }
    }
}

// ---------------------------------------------------------------------------
// Kernel B: per-edge gather from L2-resident hW, scale by norm, scatter-add.
// 64 threads per edge (one float per thread).
// ---------------------------------------------------------------------------
__global__ void __launch_bounds__(256)
rgcn_edge_kernel(const float* __restrict__ hW,
                 const float* __restrict__ norm,
                 const int* __restrict__ src,
                 const int* __restrict__ dst,
                 const int* __restrict__ et,
                 float* __restrict__ acc, int E)
{
    int gid = blockIdx.x * blockDim.x + threadIdx.x;
    int e = gid >> 6;
    int c = gid & 63;
    if (e >= E) return;
    int s = src[e], d = dst[e], r = et[e];
    float v = hW[((size_t)r * N_NODES + s) * OUT_DIM + c] * norm[e];
    unsafeAtomicAdd(acc + (size_t)d * OUT_DIM + c, v);   // global_atomic_add_f32
}

// ---------------------------------------------------------------------------
// Kernel C: final ReLU into d_out.
// ---------------------------------------------------------------------------
__global__ void __launch_bounds__(256)
rgcn_relu_kernel(const float* __restrict__ acc, float* __restrict__ out, int n)
{
    int i = blockIdx.x * blockDim.x + threadIdx.x;
    if (i < n) out[i] = fmaxf(acc[i], 0.0f);
}

// ---------------------------------------------------------------------------
extern "C" void kernel_launch(void* const* d_in, const int* in_sizes, int n_in,
                              void* d_out, int out_size, void* d_ws, size_t ws_size,
                              hipStream_t stream)
{
    const float* X     = (const float*)d_in[0];
    const float* norm  = (const float*)d_in[1];
    const float* W1    = (const float*)d_in[2];
    const float* b1    = (const float*)d_in[3];
    const float* W2    = (const float*)d_in[4];
    const float* b2    = (const float*)d_in[5];
    const float* Wrel  = (const float*)d_in[6];
    const float* Wloop = (const float*)d_in[7];
    const float* bias  = (const float*)d_in[8];
    const int*   src   = (const int*)d_in[9];
    const int*   dst   = (const int*)d_in[10];
    const int*   et    = (const int*)d_in[11];
    const int    E     = in_sizes[9];

    // Workspace layout: hW [R,N,OUT] (102.4 MB, L2-resident) then acc [N,OUT].
    float* hW  = (float*)d_ws;
    float* acc = (float*)((char*)d_ws +
                          (size_t)N_REL * N_NODES * OUT_DIM * sizeof(float));
    float* out = (float*)d_out;

    rgcn_node_kernel<<<(N_TILES + 3) / 4, 128, 0, stream>>>(
        X, W1, b1, W2, b2, Wrel, Wloop, bias, hW, acc);

    long ethreads = (long)E * 64;
    rgcn_edge_kernel<<<(int)((ethreads + 255) / 256), 256, 0, stream>>>(
        hW, norm, src, dst, et, acc, E);

    int n = N_NODES * OUT_DIM;
    rgcn_relu_kernel<<<(n + 255) / 256, 256, 0, stream>>>(acc, out, n);
}